// GAT1_72181220377181
// MI455X (gfx1250) — compile-verified
//
#include <hip/hip_runtime.h>
#include <hip/hip_bf16.h>
#include <math.h>

#define N_NODES  20000
#define N_EDGES  320000
#define N_GRAPHS 50
#define IN_DIM   64
#define NH1      10
#define ND1      64
#define NH2      1
#define ND2      128
#define NEG_SLOPE 0.2f

typedef __attribute__((ext_vector_type(2))) float v2f;
typedef __attribute__((ext_vector_type(8))) float v8f;

static inline int cdiv(int a, int b) { return (a + b - 1) / b; }

// ---------------------------------------------------------------------------
// Float atomic max via ordered-int trick (works for mixed signs, no NaNs).
// -inf init value (0xFF800000) loses to any finite value under both paths.
// ---------------------------------------------------------------------------
__device__ inline void atomicMaxF(float* addr, float val) {
  if (val >= 0.0f) {
    atomicMax((int*)addr, __float_as_int(val));
  } else {
    atomicMin((unsigned int*)addr, __float_as_uint(val));
  }
}

__global__ void fill_f32(float* p, float v, int n) {
  int i = blockIdx.x * blockDim.x + threadIdx.x;
  if (i < n) p[i] = v;
}

// ---------------------------------------------------------------------------
// C[nrows, NCOLS] = A[nrows, 64] x B[64, NCOLS], fp32 WMMA 16x16x4.
// Block = 64 threads (2 waves). Both waves share one 16-column tile of B,
// which is async-staged into LDS (transposed [col][k]) via the gfx1250
// ASYNCcnt path, then each wave computes one 16x16 output tile.
// Grid: x = NCOLS/16 column tiles, y = (nrows/16)/2 row-tile pairs (exact).
// A tile layout (ISA 7.12.2, f32 16x4): lanes 0-15 hold K={k0,k0+1},
// lanes 16-31 hold K={k0+2,k0+3}; B mirrors it; C/D = 8 VGPRs per lane.
// ---------------------------------------------------------------------------
template <int NCOLS>
__global__ void gemm_f32_wmma(const float* __restrict__ A,
                              const float* __restrict__ B,
                              float* __restrict__ C) {
  constexpr int K = 64;
  __shared__ float tileB[K * 16];  // transposed: tileB[col*K + k]

  const int wave = threadIdx.x >> 5;
  const int lane = threadIdx.x & 31;
  const int colTile = blockIdx.x;
  const int rowTile = blockIdx.y * 2 + wave;

  // ---- async-stage B[0:64, colTile*16 .. +16] into LDS, transposed ----
  {
    const float* bBase = B + colTile * 16;
#pragma unroll
    for (int i = 0; i < 16; ++i) {
      int lin = threadIdx.x + i * 64;      // 0..1023
      int k = lin >> 4, col = lin & 15;
      unsigned ldsOff = (unsigned)(size_t)&tileB[col * K + k];
      unsigned long long g = (unsigned long long)(size_t)(bBase + k * NCOLS + col);
      asm volatile("global_load_async_to_lds_b32 %0, %1, off"
                   :: "v"(ldsOff), "v"(g) : "memory");
    }
    asm volatile("s_wait_asynccnt 0x0" ::: "memory");
  }
  __syncthreads();

  const int l15  = lane & 15;
  const int kgrp = (lane >> 4) << 1;       // 0 or 2
  const float* aRow = A + (size_t)(rowTile * 16 + l15) * K + kgrp;
  const float* bCol = &tileB[l15 * K + kgrp];

  v8f acc = {};
#pragma unroll
  for (int k0 = 0; k0 < K; k0 += 4) {
    v2f a = *(const v2f*)(aRow + k0);      // global_load_b64, contiguous
    v2f b = *(const v2f*)(bCol + k0);      // ds_load_b64, contiguous
    acc = __builtin_amdgcn_wmma_f32_16x16x4_f32(false, a, false, b,
                                                (short)0, acc, false, false);
  }

  const int rbase = rowTile * 16 + (lane >> 4) * 8;  // VGPR r -> row r (+8 hi)
  float* cp = C + (size_t)rbase * NCOLS + colTile * 16 + l15;
#pragma unroll
  for (int r = 0; r < 8; ++r) cp[r * NCOLS] = acc[r];
}

// ---------------------------------------------------------------------------
// Per-(node, head) attention coefficients: el = <ft, al>, er = <ft, ar>
// ---------------------------------------------------------------------------
__global__ void attn_coef1(const float* __restrict__ ft,
                           const float* __restrict__ al,
                           const float* __restrict__ ar,
                           float* __restrict__ el, float* __restrict__ er) {
  int idx = blockIdx.x * blockDim.x + threadIdx.x;
  if (idx >= N_NODES * NH1) return;
  int hh = idx % NH1;
  const float* f  = ft + (size_t)idx * ND1;  // [n][hh*64 + d]
  const float* ap = al + hh * ND1;
  const float* bp = ar + hh * ND1;
  float sl = 0.f, sr = 0.f;
#pragma unroll 8
  for (int d = 0; d < ND1; ++d) { float v = f[d]; sl += v * ap[d]; sr += v * bp[d]; }
  el[idx] = sl; er[idx] = sr;
}

__global__ void attn_coef2(const float* __restrict__ ft,
                           const float* __restrict__ al,
                           const float* __restrict__ ar,
                           float* __restrict__ el, float* __restrict__ er) {
  int n = blockIdx.x * blockDim.x + threadIdx.x;
  if (n >= N_NODES) return;
  const float* f = ft + (size_t)n * ND2;
  float sl = 0.f, sr = 0.f;
#pragma unroll 8
  for (int d = 0; d < ND2; ++d) { float v = f[d]; sl += v * al[d]; sr += v * ar[d]; }
  el[n] = sl; er[n] = sr;
}

// ---------------------------------------------------------------------------
// Edge phase: leaky-relu score + segment max, then exp + segment sum.
// ---------------------------------------------------------------------------
template <int H>
__global__ void edge_score_max(const int* __restrict__ src, const int* __restrict__ dst,
                               const float* __restrict__ el, const float* __restrict__ er,
                               float* __restrict__ e_out, float* __restrict__ m) {
  int idx = blockIdx.x * blockDim.x + threadIdx.x;
  if (idx >= N_EDGES * H) return;
  int e = idx / H, hh = idx % H;
  float v = el[src[e] * H + hh] + er[dst[e] * H + hh];
  v = (v > 0.f) ? v : NEG_SLOPE * v;
  e_out[idx] = v;
  atomicMaxF(&m[dst[e] * H + hh], v);
}

template <int H>
__global__ void edge_exp_sum(const int* __restrict__ dst,
                             float* __restrict__ e_vals,
                             const float* __restrict__ m, float* __restrict__ s) {
  int idx = blockIdx.x * blockDim.x + threadIdx.x;
  if (idx >= N_EDGES * H) return;
  int e = idx / H, hh = idx % H;
  int t = dst[e] * H + hh;
  float mm = m[t];
  if (!(mm > -1e37f)) mm = 0.f;  // zero in-degree guard (matches isfinite fixup)
  float v = __expf(e_vals[idx] - mm);
  e_vals[idx] = v;
  atomicAdd(&s[t], v);
}

template <int H, int D>
__global__ void edge_aggregate(const int* __restrict__ src, const int* __restrict__ dst,
                               const float* __restrict__ ft,
                               const float* __restrict__ e_vals,
                               const float* __restrict__ s, float* __restrict__ out) {
  int idx = blockIdx.x * blockDim.x + threadIdx.x;
  if (idx >= N_EDGES * D) return;
  int e = idx / D, d = idx % D;
  int sn = src[e], tn = dst[e];
#pragma unroll
  for (int hh = 0; hh < H; ++hh) {
    float ssum = s[tn * H + hh];
    float a = e_vals[e * H + hh] / (ssum > 0.f ? ssum : 1.f);
    atomicAdd(&out[((size_t)tn * H + hh) * D + d],
              ft[((size_t)sn * H + hh) * D + d] * a);
  }
}

// relu(out1 + b1) then sum over the 10 heads -> h[N,64]
__global__ void relu_sum_heads(const float* __restrict__ out1,
                               const float* __restrict__ b1,
                               float* __restrict__ hout) {
  int idx = blockIdx.x * blockDim.x + threadIdx.x;
  if (idx >= N_NODES * ND1) return;
  int n = idx >> 6, d = idx & 63;
  float acc = 0.f;
#pragma unroll
  for (int hh = 0; hh < NH1; ++hh) {
    float v = out1[((size_t)n * NH1 + hh) * ND1 + d] + b1[hh * ND1 + d];
    acc += (v > 0.f) ? v : 0.f;
  }
  hout[idx] = acc;
}

// relu(out2 + b2), segment-max per graph into r[G,128]
__global__ void readout_max(const float* __restrict__ out2,
                            const float* __restrict__ b2,
                            const int* __restrict__ gid, float* __restrict__ r) {
  int idx = blockIdx.x * blockDim.x + threadIdx.x;
  if (idx >= N_NODES * ND2) return;
  int n = idx >> 7, d = idx & 127;
  float v = out2[idx] + b2[d];
  v = (v > 0.f) ? v : 0.f;
  atomicMaxF(&r[gid[n] * ND2 + d], v);
}

// Final MLP: one block per graph, 128 threads.
__global__ void mlp_readout(const float* __restrict__ r,
                            const float* __restrict__ lw1, const float* __restrict__ lb1,
                            const float* __restrict__ lw2, const float* __restrict__ lb2,
                            float* __restrict__ out) {
  __shared__ float rr[ND2];
  __shared__ float red[ND2];
  int g = blockIdx.x, j = threadIdx.x;
  float v = r[(size_t)g * ND2 + j];
  rr[j] = (v > -1e37f) ? v : 0.f;  // isfinite fixup for empty graphs
  __syncthreads();
  float acc = lb1[j];
#pragma unroll 8
  for (int k = 0; k < ND2; ++k) acc += rr[k] * lw1[k * ND2 + j];
  float t = (acc > 0.f) ? acc : 0.f;
  red[j] = t * lw2[j];
  __syncthreads();
  for (int sft = ND2 >> 1; sft > 0; sft >>= 1) {
    if (j < sft) red[j] += red[j + sft];
    __syncthreads();
  }
  if (j == 0) {
    float o = red[0] + lb2[0];
    out[g] = (o > 0.f) ? o : 0.f;
  }
}

// ---------------------------------------------------------------------------
extern "C" void kernel_launch(void* const* d_in, const int* in_sizes, int n_in,
                              void* d_out, int out_size, void* d_ws, size_t ws_size,
                              hipStream_t stream) {
  (void)in_sizes; (void)n_in; (void)out_size; (void)ws_size;
  const float* feat = (const float*)d_in[0];
  const int*   src  = (const int*)d_in[1];
  const int*   dst  = (const int*)d_in[2];
  const int*   gid  = (const int*)d_in[3];
  const float* W1   = (const float*)d_in[4];
  const float* al1  = (const float*)d_in[5];
  const float* ar1  = (const float*)d_in[6];
  const float* b1   = (const float*)d_in[7];
  const float* W2   = (const float*)d_in[8];
  const float* al2  = (const float*)d_in[9];
  const float* ar2  = (const float*)d_in[10];
  const float* b2   = (const float*)d_in[11];
  const float* lw1  = (const float*)d_in[12];
  const float* lb1  = (const float*)d_in[13];
  const float* lw2  = (const float*)d_in[14];
  const float* lb2  = (const float*)d_in[15];
  float* out = (float*)d_out;

  float* ws = (float*)d_ws;
  size_t off = 0;
  float* ft1  = ws + off; off += (size_t)N_NODES * NH1 * ND1;  // 12.8M
  float* out1 = ws + off; off += (size_t)N_NODES * NH1 * ND1;  // 12.8M
  float* ex1  = ws + off; off += (size_t)N_EDGES * NH1;        // 3.2M
  float* el1  = ws + off; off += (size_t)N_NODES * NH1;
  float* er1  = ws + off; off += (size_t)N_NODES * NH1;
  float* m1   = ws + off; off += (size_t)N_NODES * NH1;
  float* s1   = ws + off; off += (size_t)N_NODES * NH1;
  float* hbuf = ws + off; off += (size_t)N_NODES * ND1;
  float* ft2  = ws + off; off += (size_t)N_NODES * ND2;
  float* out2 = ws + off; off += (size_t)N_NODES * ND2;
  float* ex2  = ws + off; off += (size_t)N_EDGES;
  float* el2  = ws + off; off += (size_t)N_NODES;
  float* er2  = ws + off; off += (size_t)N_NODES;
  float* m2   = ws + off; off += (size_t)N_NODES;
  float* s2   = ws + off; off += (size_t)N_NODES;
  float* rbuf = ws + off; off += (size_t)N_GRAPHS * ND2;

  const float NEG_INF = -__builtin_inff();

  // ---- init ----
  hipMemsetAsync(out1, 0, (size_t)N_NODES * NH1 * ND1 * sizeof(float), stream);
  hipMemsetAsync(out2, 0, (size_t)N_NODES * ND2 * sizeof(float), stream);
  hipMemsetAsync(s1,   0, (size_t)N_NODES * NH1 * sizeof(float), stream);
  hipMemsetAsync(s2,   0, (size_t)N_NODES * sizeof(float), stream);
  fill_f32<<<cdiv(N_NODES * NH1, 256), 256, 0, stream>>>(m1, NEG_INF, N_NODES * NH1);
  fill_f32<<<cdiv(N_NODES, 256), 256, 0, stream>>>(m2, NEG_INF, N_NODES);
  fill_f32<<<cdiv(N_GRAPHS * ND2, 256), 256, 0, stream>>>(rbuf, NEG_INF, N_GRAPHS * ND2);

  // ---- layer 1 ----
  {
    dim3 grid((NH1 * ND1) / 16, (N_NODES / 16) / 2);  // (40, 625)
    gemm_f32_wmma<NH1 * ND1><<<grid, 64, 0, stream>>>(feat, W1, ft1);
  }
  attn_coef1<<<cdiv(N_NODES * NH1, 256), 256, 0, stream>>>(ft1, al1, ar1, el1, er1);
  edge_score_max<NH1><<<cdiv(N_EDGES * NH1, 256), 256, 0, stream>>>(src, dst, el1, er1, ex1, m1);
  edge_exp_sum<NH1><<<cdiv(N_EDGES * NH1, 256), 256, 0, stream>>>(dst, ex1, m1, s1);
  edge_aggregate<NH1, ND1><<<cdiv(N_EDGES * ND1, 256), 256, 0, stream>>>(src, dst, ft1, ex1, s1, out1);
  relu_sum_heads<<<cdiv(N_NODES * ND1, 256), 256, 0, stream>>>(out1, b1, hbuf);

  // ---- layer 2 ----
  {
    dim3 grid(ND2 / 16, (N_NODES / 16) / 2);  // (8, 625)
    gemm_f32_wmma<ND2><<<grid, 64, 0, stream>>>(hbuf, W2, ft2);
  }
  attn_coef2<<<cdiv(N_NODES, 256), 256, 0, stream>>>(ft2, al2, ar2, el2, er2);
  edge_score_max<NH2><<<cdiv(N_EDGES, 256), 256, 0, stream>>>(src, dst, el2, er2, ex2, m2);
  edge_exp_sum<NH2><<<cdiv(N_EDGES, 256), 256, 0, stream>>>(dst, ex2, m2, s2);
  edge_aggregate<NH2, ND2><<<cdiv(N_EDGES * ND2, 256), 256, 0, stream>>>(src, dst, ft2, ex2, s2, out2);

  // ---- readout ----
  readout_max<<<cdiv(N_NODES * ND2, 256), 256, 0, stream>>>(out2, b2, gid, rbuf);
  mlp_readout<<<N_GRAPHS, ND2, 0, stream>>>(rbuf, lw1, lb1, lw2, lb2, out);
}